// SMPLXLayer_37649683317222
// MI455X (gfx1250) — compile-verified
//
#include <hip/hip_runtime.h>
#include <math.h>

typedef __attribute__((ext_vector_type(2))) float v2f;
typedef __attribute__((ext_vector_type(8))) float v8f;

#define BATCH 1024
#define NVERT 10475
#define NV3   31425          // V*3
#define NJ    55
#define KPOSE 486            // (J-1)*9
#define PFSTR 488            // padded pose-feature stride (8B-aligned pairs)
#define RELSTR 896           // 56*16 padded rel tile per batch
#define NT64  492            // ceil(NV3/64) column groups for k_vposed

__constant__ int c_parents[NJ] = {
  -1,0,0,0,1,2,3,4,5,6,7,8,9,9,9,12,13,14,16,17,18,19,15,15,15,
  20,25,26,20,28,29,20,31,32,20,34,35,20,37,38,21,40,41,21,43,44,
  21,46,47,21,49,50,21,52,53};

// ---------------------------------------------------------------------------
// K1: fold J_regressor into tiny tables:
//   Jws[(j*3+c)*11 + l]  = sum_v Jreg[j,v] * shapedirs[v,c,l]   (l<10)
//   Jws[(j*3+c)*11 + 10] = sum_v Jreg[j,v] * v_template[v,c]
// ---------------------------------------------------------------------------
__global__ __launch_bounds__(256) void k_jreg(
    const float* __restrict__ Jreg, const float* __restrict__ v_template,
    const float* __restrict__ shapedirs, float* __restrict__ Jws)
{
  __shared__ float red[256];
  const int o = blockIdx.x;            // 0..1814
  const int j = o / 33;
  const int rem = o % 33;
  const int c = rem / 11;
  const int l = rem % 11;
  const float* Jr = Jreg + j * NVERT;
  float acc = 0.f;
  for (int v = threadIdx.x; v < NVERT; v += 256) {
    float w = Jr[v];
    float x = (l < 10) ? shapedirs[(v*3 + c)*10 + l] : v_template[v*3 + c];
    acc += w * x;
  }
  red[threadIdx.x] = acc;
  __syncthreads();
  for (int s = 128; s > 0; s >>= 1) {
    if (threadIdx.x < s) red[threadIdx.x] += red[threadIdx.x + s];
    __syncthreads();
  }
  if (threadIdx.x == 0) Jws[(j*3 + c)*11 + l] = red[0];
}

// ---------------------------------------------------------------------------
// K2: per-batch Rodrigues + joints + kinematic chain -> rel (56x16 padded)
//     and pose_feature (486, stride 488)
// ---------------------------------------------------------------------------
__global__ __launch_bounds__(64) void k_transforms(
    const float* __restrict__ orient, const float* __restrict__ body_pose,
    const float* __restrict__ betas, const float* __restrict__ Jws,
    float* __restrict__ relws, float* __restrict__ pfws)
{
  __shared__ float sR[NJ][9];
  __shared__ float sJ[NJ][3];
  __shared__ float sG[NJ][12];   // world transforms, 3x4 row-major
  const int b = blockIdx.x;
  const int tid = threadIdx.x;

  if (tid < NJ) {
    float rx, ry, rz;
    if (tid == 0) { rx = orient[b*3]; ry = orient[b*3+1]; rz = orient[b*3+2]; }
    else { const float* p = body_pose + b*162 + (tid-1)*3; rx = p[0]; ry = p[1]; rz = p[2]; }
    rx += 1e-8f; ry += 1e-8f; rz += 1e-8f;
    float ang = sqrtf(rx*rx + ry*ry + rz*rz);
    float inv = 1.0f / ang;
    float x = rx*inv, y = ry*inv, z = rz*inv;
    float s = sinf(ang), co = cosf(ang), t = 1.0f - co;
    sR[tid][0] = 1.f - t*(y*y + z*z);
    sR[tid][1] = -s*z + t*x*y;
    sR[tid][2] =  s*y + t*x*z;
    sR[tid][3] =  s*z + t*x*y;
    sR[tid][4] = 1.f - t*(x*x + z*z);
    sR[tid][5] = -s*x + t*y*z;
    sR[tid][6] = -s*y + t*x*z;
    sR[tid][7] =  s*x + t*y*z;
    sR[tid][8] = 1.f - t*(x*x + y*y);
    const float* bb = betas + b*10;
#pragma unroll
    for (int c = 0; c < 3; ++c) {
      const float* Jr = Jws + (tid*3 + c)*11;
      float v = Jr[10];
#pragma unroll
      for (int l = 0; l < 10; ++l) v += Jr[l] * bb[l];
      sJ[tid][c] = v;
    }
  }
  __syncthreads();

  if (tid == 0) {
#pragma unroll
    for (int m = 0; m < 3; ++m) {
      sG[0][m*4+0] = sR[0][m*3+0];
      sG[0][m*4+1] = sR[0][m*3+1];
      sG[0][m*4+2] = sR[0][m*3+2];
      sG[0][m*4+3] = sJ[0][m];
    }
    for (int j = 1; j < NJ; ++j) {
      int p = c_parents[j];
      float lt0 = sJ[j][0]-sJ[p][0], lt1 = sJ[j][1]-sJ[p][1], lt2 = sJ[j][2]-sJ[p][2];
      for (int m = 0; m < 3; ++m) {
        float g0 = sG[p][m*4+0], g1 = sG[p][m*4+1], g2 = sG[p][m*4+2];
        sG[j][m*4+0] = g0*sR[j][0] + g1*sR[j][3] + g2*sR[j][6];
        sG[j][m*4+1] = g0*sR[j][1] + g1*sR[j][4] + g2*sR[j][7];
        sG[j][m*4+2] = g0*sR[j][2] + g1*sR[j][5] + g2*sR[j][8];
        sG[j][m*4+3] = g0*lt0 + g1*lt1 + g2*lt2 + sG[p][m*4+3];
      }
    }
  }
  __syncthreads();

  float* relb = relws + (size_t)b * RELSTR;
  if (tid < NJ) {
#pragma unroll
    for (int m = 0; m < 3; ++m) {
      float r0 = sG[tid][m*4+0], r1 = sG[tid][m*4+1], r2 = sG[tid][m*4+2];
      relb[tid*16 + m*4+0] = r0;
      relb[tid*16 + m*4+1] = r1;
      relb[tid*16 + m*4+2] = r2;
      relb[tid*16 + m*4+3] = sG[tid][m*4+3]
          - (r0*sJ[tid][0] + r1*sJ[tid][1] + r2*sJ[tid][2]);
    }
    relb[tid*16+12] = 0.f; relb[tid*16+13] = 0.f;
    relb[tid*16+14] = 0.f; relb[tid*16+15] = 0.f;
  }
  if (tid == NJ) {                      // zero K-padding row 55
#pragma unroll
    for (int n = 0; n < 16; ++n) relb[NJ*16 + n] = 0.f;
  }
  for (int pp = tid; pp < KPOSE; pp += 64) {
    int j = pp / 9 + 1;
    int e = pp % 9;
    float diag = (e == 0 || e == 4 || e == 8) ? 1.f : 0.f;
    pfws[(size_t)b*PFSTR + pp] = sR[j][e] - diag;
  }
}

// ---------------------------------------------------------------------------
// K3: v_posed GEMM, M=1024 (batch), N=31425, K=486 via v_wmma_f32_16x16x4.
// Each wave computes a 16x64 tile (4 accumulators). Single-basic-block
// K-loop (#pragma unroll 1, affine pointer bumps) so accumulators stay
// pinned in place (no phi copies); speculative global_prefetch streams
// posedirs 2 K-steps ahead. Epilogue adds k=484,485 + v_template +
// betas*shapedirs.
// ---------------------------------------------------------------------------
__global__ __launch_bounds__(256) void k_vposed(
    const float* __restrict__ pf, const float* __restrict__ posedirs,
    const float* __restrict__ betas, const float* __restrict__ shapedirs,
    const float* __restrict__ v_template, float* __restrict__ vposed)
{
  const int lane = threadIdx.x & 31;
  const int tile = blockIdx.x * 8 + (threadIdx.x >> 5);   // tm-major order
  const int tm   = tile / NT64;                           // batch tile 0..63
  const int tn   = tile % NT64;                           // 64-col group
  const int half = (lane < 16) ? 0 : 2;
  const int rowA = tm * 16 + (lane & 15);                 // batch row (A M)
  const int colB = tn * 64 + (lane & 15);                 // first column (N)

  const float* aptr = pf + (size_t)rowA * PFSTR + half;       // A[M][ka..ka+1]
  const float* bptr = posedirs + (size_t)half * NV3 + colB;   // B[ka][N..]
  const float* pfe  = posedirs + (size_t)8 * NV3 + colB;      // prefetch ptr

  v8f acc[4];
#pragma unroll
  for (int t = 0; t < 4; ++t) acc[t] = (v8f){0.f,0.f,0.f,0.f,0.f,0.f,0.f,0.f};

#pragma unroll 1
  for (int k0 = 0; k0 < 484; k0 += 4) {
    // speculative prefetch (TH=RT speculative: OOB rows silently dropped)
    __builtin_prefetch(pfe, 0, 3);
    __builtin_prefetch(pfe + 32, 0, 3);
    v2f a = *(const v2f*)aptr;
#pragma unroll
    for (int t = 0; t < 4; ++t) {
      v2f bv;
      bv.x = bptr[16*t];                       // B[ka][colB+16t]
      bv.y = bptr[(size_t)NV3 + 16*t];         // B[ka+1][colB+16t]
      acc[t] = __builtin_amdgcn_wmma_f32_16x16x4_f32(false, a, false, bv,
                                                     (short)0, acc[t],
                                                     false, false);
    }
    aptr += 4;
    bptr += (size_t)4 * NV3;
    pfe  += (size_t)4 * NV3;
  }

#pragma unroll
  for (int t = 0; t < 4; ++t) {
    const int colt = colB + 16*t;
    if (colt < NV3) {
      const float b484 = posedirs[(size_t)484 * NV3 + colt];
      const float b485 = posedirs[(size_t)485 * NV3 + colt];
      const float vt = v_template[colt];
      float sd[10];
#pragma unroll
      for (int l = 0; l < 10; ++l) sd[l] = shapedirs[colt*10 + l];
#pragma unroll
      for (int r = 0; r < 8; ++r) {
        const int brow = tm*16 + ((lane < 16) ? r : r + 8); // D row -> batch
        const float* pfb = pf + (size_t)brow * PFSTR;
        float v = acc[t][r] + vt + pfb[484]*b484 + pfb[485]*b485;
        const float* bb = betas + brow*10;
#pragma unroll
        for (int l = 0; l < 10; ++l) v += bb[l]*sd[l];
        vposed[(size_t)brow * NV3 + colt] = v;
      }
    }
  }
}

// ---------------------------------------------------------------------------
// K4: skinning. Per (batch, 16-vertex tile): C = W(16x56) @ rel(56x16)
// via 14 straight-line v_wmma_f32_16x16x4 (fully unrolled K); 8 tiles per
// 256-thread block; regroup C through LDS; combine with v_posed, add
// transl, store verts.
// ---------------------------------------------------------------------------
__global__ __launch_bounds__(256) void k_skin(
    const float* __restrict__ lbs, const float* __restrict__ relws,
    const float* __restrict__ vposed, const float* __restrict__ transl,
    float* __restrict__ out)
{
  __shared__ float sC[8][256];
  const int wave = threadIdx.x >> 5;
  const int lane = threadIdx.x & 31;
  int tv = blockIdx.x * 8 + wave;            // vertex tile
  if (tv > 654) tv = 654;                    // clamp: duplicate identical work
  const int b  = blockIdx.y;                 // batch
  const int half = (lane < 16) ? 0 : 2;
  const int vrow = tv*16 + (lane & 15);      // vertex row (A frag M)
  const int coln = lane & 15;                // (m*4+n) column
  const bool vok = (vrow < NVERT);
  const float* relb = relws + (size_t)b * RELSTR;
  const float* wrow = lbs + (size_t)(vok ? vrow : 0) * NJ;

  v8f c = {0.f,0.f,0.f,0.f,0.f,0.f,0.f,0.f};
#pragma unroll
  for (int k0 = 0; k0 < 56; k0 += 4) {
    const int ka = k0 + half;
    v2f a, bv;
    a.x = (vok && ka     < NJ) ? wrow[ka]     : 0.f;
    a.y = (vok && ka + 1 < NJ) ? wrow[ka + 1] : 0.f;
    bv.x = relb[ka*16 + coln];                // row 55 is zero-padded
    bv.y = relb[(ka + 1)*16 + coln];
    c = __builtin_amdgcn_wmma_f32_16x16x4_f32(false, a, false, bv,
                                              (short)0, c, false, false);
  }

  const int mbase = (lane < 16) ? 0 : 8;
#pragma unroll
  for (int r = 0; r < 8; ++r) sC[wave][(mbase + r)*16 + coln] = c[r];
  __syncthreads();

  for (int idx = lane; idx < 48; idx += 32) {
    const int vl = idx & 15;
    const int m  = idx >> 4;
    const int v  = tv*16 + vl;
    if (v < NVERT) {
      const float* vp = vposed + (size_t)b * NV3 + v*3;
      const float* t  = sC[wave] + vl*16 + m*4;
      float val = t[0]*vp[0] + t[1]*vp[1] + t[2]*vp[2] + t[3] + transl[b*3 + m];
      out[((size_t)b * NVERT + v)*3 + m] = val;
    }
  }
}

// ---------------------------------------------------------------------------
extern "C" void kernel_launch(void* const* d_in, const int* in_sizes, int n_in,
                              void* d_out, int out_size, void* d_ws, size_t ws_size,
                              hipStream_t stream) {
  const float* transl     = (const float*)d_in[0];
  const float* orient     = (const float*)d_in[1];
  const float* betas      = (const float*)d_in[2];
  const float* body_pose  = (const float*)d_in[3];
  const float* v_template = (const float*)d_in[4];
  const float* shapedirs  = (const float*)d_in[5];
  const float* posedirs   = (const float*)d_in[6];
  const float* Jreg       = (const float*)d_in[7];
  const float* lbs        = (const float*)d_in[8];
  // d_in[9] (parents) is the fixed SMPLX constant; baked into __constant__.

  float* ws    = (float*)d_ws;
  float* Jws   = ws;                                   // 1815 (padded 1816)
  float* pfws  = ws + 1816;                            // 1024*488
  float* relws = pfws + (size_t)BATCH * PFSTR;         // 1024*896
  float* vpws  = relws + (size_t)BATCH * RELSTR;       // 1024*31425
  float* outp  = (float*)d_out;

  k_jreg      <<<1815, 256, 0, stream>>>(Jreg, v_template, shapedirs, Jws);
  k_transforms<<<BATCH, 64, 0, stream>>>(orient, body_pose, betas, Jws, relws, pfws);
  k_vposed    <<<(64 * NT64) / 8, 256, 0, stream>>>(pfws, posedirs, betas,
                                                    shapedirs, v_template, vpws);
  k_skin      <<<dim3(82, BATCH), 256, 0, stream>>>(lbs, relws, vpws, transl, outp);
}